// TLN_13589276524997
// MI455X (gfx1250) — compile-verified
//
#include <hip/hip_runtime.h>
#include <math.h>

typedef __attribute__((ext_vector_type(2))) float v2f;
typedef __attribute__((ext_vector_type(8))) float v8f;

#define TPB 256
static inline int cdiv(int a, int b) { return (a + b - 1) / b; }

__device__ __forceinline__ float sigm(float x) { return 1.0f / (1.0f + expf(-x)); }

// ---------------- utility ----------------
__global__ void k_zero(float* __restrict__ p, int n) {
  int i = blockIdx.x * blockDim.x + threadIdx.x;
  if (i < n) p[i] = 0.f;
}

// pad cls_w1 (1024 x 170) -> (1024 x 176), zero cols 170..175
__global__ void k_padw1(const float* __restrict__ w, float* __restrict__ o) {
  int i = blockIdx.x * blockDim.x + threadIdx.x;
  if (i >= 1024 * 176) return;
  int col = i % 176, row = i / 176;
  o[i] = (col < 170) ? w[row * 170 + col] : 0.f;
}

// pad cls_w3 (10 x 256) -> (16 x 256), zero rows 10..15
__global__ void k_padw3(const float* __restrict__ w, float* __restrict__ o) {
  int i = blockIdx.x * blockDim.x + threadIdx.x;
  if (i >= 16 * 256) return;
  int row = i / 256;
  o[i] = (row < 10) ? w[i] : 0.f;
}

// ---------------- stage 1: conv1d(1->16, k=32) ----------------
__global__ void k_conv1(const float* __restrict__ x, const float* __restrict__ w,
                        const float* __restrict__ b, float* __restrict__ f0) {
  const int L = 687;
  int i = blockIdx.x * blockDim.x + threadIdx.x;
  if (i >= 64 * 16 * L) return;
  int l = i % L, c = (i / L) % 16, bb = i / (L * 16);
  const float* xr = x + bb * 718 + l;
  const float* wr = w + c * 32;
  float acc = b[c];
#pragma unroll
  for (int k = 0; k < 32; ++k) acc += xr[k] * wr[k];
  f0[i] = acc;
}

// ---------------- stage 2: batchnorm stats (one block per channel) ----------------
__global__ void k_bnstats(const float* __restrict__ f0, float* __restrict__ mean,
                          float* __restrict__ rstd) {
  __shared__ float s1[TPB], s2[TPB];
  const int L = 687, N = 64 * L;
  int c = blockIdx.x, tid = threadIdx.x;
  float su = 0.f, sq = 0.f;
  for (int j = tid; j < N; j += TPB) {
    int bb = j / L, l = j % L;
    float v = f0[(bb * 16 + c) * L + l];
    su += v; sq += v * v;
  }
  s1[tid] = su; s2[tid] = sq; __syncthreads();
  for (int st = TPB / 2; st > 0; st >>= 1) {
    if (tid < st) { s1[tid] += s1[tid + st]; s2[tid] += s2[tid + st]; }
    __syncthreads();
  }
  if (tid == 0) {
    float m = s1[0] / (float)N;
    float v = s2[0] / (float)N - m * m;
    mean[c] = m; rstd[c] = rsqrtf(v + 1e-5f);
  }
}

// ---------------- stage 3: bn + relu + maxpool(3,2): 687 -> 343 ----------------
__global__ void k_bnpool(const float* __restrict__ f0, const float* __restrict__ mean,
                         const float* __restrict__ rstd, const float* __restrict__ g,
                         const float* __restrict__ be, float* __restrict__ f1) {
  int i = blockIdx.x * blockDim.x + threadIdx.x;
  if (i >= 64 * 16 * 343) return;
  int l = i % 343, c = (i / 343) % 16, b = i / (343 * 16);
  const float* row = f0 + (b * 16 + c) * 687;
  float m = mean[c], r = rstd[c], gg = g[c], bb = be[c];
  float best = -1e30f;
#pragma unroll
  for (int j = 0; j < 3; ++j) {
    float v = fmaxf(gg * (row[2 * l + j] - m) * r + bb, 0.f);
    best = fmaxf(best, v);
  }
  f1[i] = best;
}

// ---------------- stage 4: pos/neg sigmoid gates + maxpool(3,2): 343 -> 171 ----------------
__global__ void k_miu(const float* __restrict__ f1, const float* __restrict__ a_pos,
                      const float* __restrict__ b_pos, const float* __restrict__ a_neg,
                      const float* __restrict__ b_neg, float* __restrict__ x_miu) {
  int i = blockIdx.x * blockDim.x + threadIdx.x;
  if (i >= 64 * 32 * 171) return;
  int l = i % 171, c = (i / 171) % 32, b = i / (171 * 32);
  int cc = c & 15;
  bool neg = c >= 16;
  const float* row = f1 + (b * 16 + cc) * 343;
  float a = neg ? -a_neg[cc] : a_pos[cc];
  float bp = neg ? b_neg[cc] : b_pos[cc];
  float best = -1e30f;
#pragma unroll
  for (int j = 0; j < 3; ++j) best = fmaxf(best, sigm(a * (row[2 * l + j] - bp)));
  x_miu[i] = best;
}

// ---------------- conv k=3 pad=1 + act + maxpool(3,2) ----------------
__global__ void k_conv3_pool(const float* __restrict__ in, const float* __restrict__ w,
                             const float* __restrict__ bias, float* __restrict__ out,
                             int Cin, int Cout, int Lin, int Lout, int act /*0=tanh,1=relu*/) {
  int i = blockIdx.x * blockDim.x + threadIdx.x;
  if (i >= 64 * Cout * Lout) return;
  int l = i % Lout, co = (i / Lout) % Cout, b = i / (Lout * Cout);
  const float* ib = in + b * Cin * Lin;
  const float* wr = w + co * Cin * 3;
  float best = -1e30f;
  for (int j = 0; j < 3; ++j) {
    int p = 2 * l + j;
    float acc = bias[co];
    for (int ci = 0; ci < Cin; ++ci) {
      const float* r = ib + ci * Lin;
      const float* wc = wr + ci * 3;
#pragma unroll
      for (int t = 0; t < 3; ++t) {
        int q = p - 1 + t;
        if ((unsigned)q < (unsigned)Lin) acc += r[q] * wc[t];
      }
    }
    float v = act ? fmaxf(acc, 0.f) : tanhf(acc);
    best = fmaxf(best, v);
  }
  out[i] = best;
}

// ---------------- upsample2x_linear + conv k=3 pad=(2,2) + act ----------------
__device__ __forceinline__ float up_get(const float* __restrict__ r, int L, int i) {
  float src = (i + 0.5f) * 0.5f - 0.5f;
  src = fminf(fmaxf(src, 0.f), (float)(L - 1));
  int lo = (int)floorf(src);
  int hi = min(lo + 1, L - 1);
  float fr = src - (float)lo;
  return r[lo] * (1.f - fr) + r[hi] * fr;
}

__global__ void k_upconv(const float* __restrict__ in, const float* __restrict__ w,
                         const float* __restrict__ bias, float* __restrict__ out,
                         int Cin, int Cout, int Lin, int act /*0=tanh,1=sigmoid*/) {
  int Lup = 2 * Lin, Lout = Lup + 2;
  int i = blockIdx.x * blockDim.x + threadIdx.x;
  if (i >= 64 * Cout * Lout) return;
  int l = i % Lout, co = (i / Lout) % Cout, b = i / (Lout * Cout);
  const float* ib = in + b * Cin * Lin;
  const float* wr = w + co * Cin * 3;
  float acc = bias[co];
  for (int ci = 0; ci < Cin; ++ci) {
    const float* r = ib + ci * Lin;
    const float* wc = wr + ci * 3;
#pragma unroll
    for (int t = 0; t < 3; ++t) {
      int q = l - 2 + t;
      if ((unsigned)q < (unsigned)Lup) acc += up_get(r, Lin, q) * wc[t];
    }
  }
  out[i] = act ? sigm(acc) : tanhf(acc);
}

// ---------------- w = maxpool(h,4,1); segment masks; inp = sel * x_miu ----------------
__global__ void k_wsel(const float* __restrict__ hd, const float* __restrict__ x_miu,
                       float* __restrict__ inp) {
  int i = blockIdx.x * blockDim.x + threadIdx.x;
  if (i >= 64 * 32 * 171) return;
  int l = i % 171, c = (i / 171) % 32, b = i / (171 * 32);
  const float* r = hd + (b * 32 + c) * 174;
  float wv = fmaxf(fmaxf(r[l], r[l + 1]), fmaxf(r[l + 2], r[l + 3]));
  float xm = x_miu[i];
#pragma unroll
  for (int s = 0; s < 5; ++s) {
    float lo = 0.2f * s, hi = lo + 0.2f;
    bool m = (wv >= lo) && (wv < hi);
    inp[(b * 160 + s * 32 + c) * 171 + l] = m ? wv * xm : 0.f;
  }
}

// ---------------- until_op (T=35) -> x1 channels [40,120) ----------------
__global__ void k_until(const float* __restrict__ inp, float* __restrict__ x1) {
  int i = blockIdx.x * blockDim.x + threadIdx.x;
  if (i >= 64 * 80 * 170) return;
  int l = i % 170, p = (i / 170) % 80, b = i / (170 * 80);
  const float* phi1 = inp + (b * 160 + 2 * p) * 171;
  const float* phi2 = inp + (b * 160 + 2 * p + 1) * 171;
  float runmin = 1e30f, best = -1e30f;
  for (int t = 0; t < 35; ++t) {
    int idx = l + t;
    float p1, p2;
    if (idx < 171) { p1 = phi1[idx]; p2 = phi2[idx]; }
    else { p1 = 1e30f; p2 = -1e30f; }
    runmin = fminf(runmin, p1);
    best = fmaxf(best, fminf(runmin, p2));
  }
  x1[(b * 120 + 40 + p) * 170 + l] = best;
}

// ---------------- conv k=2 no pad on inp (160ch), and/or -> x1 ch [0,40) ----------------
__global__ void k_andor1(const float* __restrict__ inp, const float* __restrict__ wa,
                         const float* __restrict__ ba, const float* __restrict__ wo,
                         const float* __restrict__ bo, float* __restrict__ x1) {
  int i = blockIdx.x * blockDim.x + threadIdx.x;
  if (i >= 64 * 40 * 170) return;
  int l = i % 170, co = (i / 170) % 40, b = i / (170 * 40);
  int c = co % 20;
  const float* w = (co < 20) ? wa : wo;
  const float* bs = (co < 20) ? ba : bo;
  const float* ib = inp + b * 160 * 171;
  const float* wr = w + c * 320;
  float acc = bs[c];
  for (int ci = 0; ci < 160; ++ci) {
    const float* r = ib + ci * 171 + l;
    acc += r[0] * wr[2 * ci] + r[1] * wr[2 * ci + 1];
  }
  x1[(b * 120 + co) * 170 + l] = fmaxf(acc, 0.f);
}

// ---------------- conv k=2 pad(0,1) on x1 (120ch), and/or -> x2pad (768 x 176) ----------------
__global__ void k_andor2(const float* __restrict__ x1, const float* __restrict__ wa,
                         const float* __restrict__ ba, const float* __restrict__ wo,
                         const float* __restrict__ bo, float* __restrict__ x2pad) {
  int i = blockIdx.x * blockDim.x + threadIdx.x;
  if (i >= 64 * 12 * 170) return;
  int l = i % 170, c = (i / 170) % 12, b = i / (170 * 12);
  int cc = c % 6;
  const float* w = (c < 6) ? wa : wo;
  const float* bs = (c < 6) ? ba : bo;
  const float* ib = x1 + b * 120 * 170;
  const float* wr = w + cc * 240;
  float acc = bs[cc];
  for (int ci = 0; ci < 120; ++ci) {
    const float* r = ib + ci * 170;
    float v0 = r[l];
    float v1 = (l + 1 < 170) ? r[l + 1] : 0.f;
    acc += v0 * wr[2 * ci] + v1 * wr[2 * ci + 1];
  }
  x2pad[(b * 12 + c) * 176 + l] = fmaxf(acc, 0.f);
}

// ---------------- WMMA fp32 GEMM: C[M,N] = act(A[M,K] * B[K,N] + bias) ----------------
// A row-major (lda = Kpad), B[k,n] = Bw[n*ldb + k] (weight rows are N, cols K).
// One wave (32 lanes) computes one 16x16 tile via V_WMMA_F32_16X16X4_F32.
// Two interleaved accumulators break the WMMA->WMMA dependency chain.
__global__ void k_gemm_wmma(const float* __restrict__ A, int lda,
                            const float* __restrict__ Bw, int ldb,
                            const float* __restrict__ bias,
                            float* __restrict__ C, int ldc, int nstore,
                            int Kpad, int relu) {
  int m0 = blockIdx.x * 16;
  int n0 = blockIdx.y * 16;
  int lane = threadIdx.x;
  int half = lane >> 4;   // selects K pair {0,1} vs {2,3} within the 16x4 fragment
  int mr = lane & 15;     // A row within tile / B column within tile
  const float* Ar = A + (size_t)(m0 + mr) * lda + 2 * half;
  const float* Br = Bw + (size_t)(n0 + mr) * ldb + 2 * half;
  v8f acc0 = {};
  v8f acc1 = {};
  for (int k = 0; k < Kpad; k += 8) {
    v2f a0, b0, a1, b1;
    a0.x = Ar[k];     a0.y = Ar[k + 1];
    b0.x = Br[k];     b0.y = Br[k + 1];
    a1.x = Ar[k + 4]; a1.y = Ar[k + 5];
    b1.x = Br[k + 4]; b1.y = Br[k + 5];
    acc0 = __builtin_amdgcn_wmma_f32_16x16x4_f32(false, a0, false, b0,
                                                 (short)0, acc0, false, false);
    acc1 = __builtin_amdgcn_wmma_f32_16x16x4_f32(false, a1, false, b1,
                                                 (short)0, acc1, false, false);
  }
  int col = n0 + mr;
  if (col < nstore) {
    float bv = bias[col];
    int rowbase = m0 + 8 * half;   // C layout: VGPR r -> row r (+8 for lanes 16-31)
#pragma unroll
    for (int r = 0; r < 8; ++r) {
      float v = acc0[r] + acc1[r] + bv;
      if (relu) v = fmaxf(v, 0.f);
      C[(size_t)(rowbase + r) * ldc + col] = v;
    }
  }
}

// ---------------- host orchestration ----------------
extern "C" void kernel_launch(void* const* d_in, const int* in_sizes, int n_in,
                              void* d_out, int out_size, void* d_ws, size_t ws_size,
                              hipStream_t stream) {
  const float* x      = (const float*)d_in[0];
  const float* bpn_w  = (const float*)d_in[1];
  const float* bpn_b  = (const float*)d_in[2];
  const float* bn_g   = (const float*)d_in[3];
  const float* bn_b   = (const float*)d_in[4];
  const float* a_pos  = (const float*)d_in[5];
  const float* b_pos  = (const float*)d_in[6];
  const float* a_neg  = (const float*)d_in[7];
  const float* b_neg  = (const float*)d_in[8];
  const float* ae_w1  = (const float*)d_in[9];
  const float* ae_b1  = (const float*)d_in[10];
  const float* ae_w2  = (const float*)d_in[11];
  const float* ae_b2  = (const float*)d_in[12];
  const float* ae_w3  = (const float*)d_in[13];
  const float* ae_b3  = (const float*)d_in[14];
  const float* ae_w4  = (const float*)d_in[15];
  const float* ae_b4  = (const float*)d_in[16];
  const float* and2d_w = (const float*)d_in[17];
  const float* and2d_b = (const float*)d_in[18];
  const float* or2d_w  = (const float*)d_in[19];
  const float* or2d_b  = (const float*)d_in[20];
  const float* and_w   = (const float*)d_in[21];
  const float* and_b   = (const float*)d_in[22];
  const float* or_w    = (const float*)d_in[23];
  const float* or_b    = (const float*)d_in[24];
  const float* cls_w1  = (const float*)d_in[25];
  const float* cls_b1  = (const float*)d_in[26];
  const float* cls_w2  = (const float*)d_in[27];
  const float* cls_b2  = (const float*)d_in[28];
  const float* cls_w3  = (const float*)d_in[29];
  const float* cls_b3  = (const float*)d_in[30];
  float* out = (float*)d_out;

  float* ws = (float*)d_ws;
  size_t off = 0;
  auto alloc = [&](size_t n) { size_t o = off; off += (n + 63) & ~(size_t)63; return o; };
  float* f0    = ws + alloc(64 * 16 * 687);
  float* mean  = ws + alloc(16);
  float* rstd  = ws + alloc(16);
  float* f1    = ws + alloc(64 * 16 * 343);
  float* x_miu = ws + alloc(64 * 32 * 171);
  float* h_a   = ws + alloc(64 * 64 * 85);
  float* h_b   = ws + alloc(64 * 32 * 42);
  float* h_c   = ws + alloc(64 * 64 * 86);
  float* h_d   = ws + alloc(64 * 32 * 174);
  float* inp   = ws + alloc(64 * 160 * 171);
  float* x1    = ws + alloc(64 * 120 * 170);
  float* x2pad = ws + alloc(768 * 176);
  float* w1pad = ws + alloc(1024 * 176);
  float* w3pad = ws + alloc(16 * 256);
  float* h1    = ws + alloc(768 * 1024);
  float* h2    = ws + alloc(768 * 256);
  (void)ws_size; (void)n_in; (void)in_sizes; (void)out_size;

  // padded-weight / padded-activation prep
  k_zero<<<cdiv(768 * 176, TPB), TPB, 0, stream>>>(x2pad, 768 * 176);
  k_padw1<<<cdiv(1024 * 176, TPB), TPB, 0, stream>>>(cls_w1, w1pad);
  k_padw3<<<cdiv(16 * 256, TPB), TPB, 0, stream>>>(cls_w3, w3pad);

  // front-end pipeline
  k_conv1<<<cdiv(64 * 16 * 687, TPB), TPB, 0, stream>>>(x, bpn_w, bpn_b, f0);
  k_bnstats<<<16, TPB, 0, stream>>>(f0, mean, rstd);
  k_bnpool<<<cdiv(64 * 16 * 343, TPB), TPB, 0, stream>>>(f0, mean, rstd, bn_g, bn_b, f1);
  k_miu<<<cdiv(64 * 32 * 171, TPB), TPB, 0, stream>>>(f1, a_pos, b_pos, a_neg, b_neg, x_miu);

  // autoencoder
  k_conv3_pool<<<cdiv(64 * 64 * 85, TPB), TPB, 0, stream>>>(x_miu, ae_w1, ae_b1, h_a,
                                                            32, 64, 171, 85, /*tanh*/0);
  k_conv3_pool<<<cdiv(64 * 32 * 42, TPB), TPB, 0, stream>>>(h_a, ae_w2, ae_b2, h_b,
                                                            64, 32, 85, 42, /*relu*/1);
  k_upconv<<<cdiv(64 * 64 * 86, TPB), TPB, 0, stream>>>(h_b, ae_w3, ae_b3, h_c,
                                                        32, 64, 42, /*tanh*/0);
  k_upconv<<<cdiv(64 * 32 * 174, TPB), TPB, 0, stream>>>(h_c, ae_w4, ae_b4, h_d,
                                                         64, 32, 86, /*sigmoid*/1);

  // segment select + temporal-logic layers
  k_wsel<<<cdiv(64 * 32 * 171, TPB), TPB, 0, stream>>>(h_d, x_miu, inp);
  k_until<<<cdiv(64 * 80 * 170, TPB), TPB, 0, stream>>>(inp, x1);
  k_andor1<<<cdiv(64 * 40 * 170, TPB), TPB, 0, stream>>>(inp, and2d_w, and2d_b,
                                                         or2d_w, or2d_b, x1);
  k_andor2<<<cdiv(64 * 12 * 170, TPB), TPB, 0, stream>>>(x1, and_w, and_b,
                                                         or_w, or_b, x2pad);

  // classifier: three WMMA GEMMs (M=768)
  {
    dim3 g(768 / 16, 1024 / 16);
    k_gemm_wmma<<<g, 32, 0, stream>>>(x2pad, 176, w1pad, 176, cls_b1,
                                      h1, 1024, 1024, 176, /*relu*/1);
  }
  {
    dim3 g(768 / 16, 256 / 16);
    k_gemm_wmma<<<g, 32, 0, stream>>>(h1, 1024, cls_w2, 1024, cls_b2,
                                      h2, 256, 256, 1024, /*relu*/1);
  }
  {
    dim3 g(768 / 16, 1);
    k_gemm_wmma<<<g, 32, 0, stream>>>(h2, 256, w3pad, 256, cls_b3,
                                      out, 10, 10, 256, /*none*/0);
  }
}